// get_model_28956669510155
// MI455X (gfx1250) — compile-verified
//
#include <hip/hip_runtime.h>
#include <hip/hip_bf16.h>
#include <math.h>

typedef __attribute__((ext_vector_type(2))) float v2f;
typedef __attribute__((ext_vector_type(8))) float v8f;

#define BN_EPS 1e-5f

static inline unsigned grid1(size_t n, unsigned b = 256) {
  return (unsigned)((n + b - 1) / b);
}
static inline int cdiv(int a, int b) { return (a + b - 1) / b; }

// ---------------------------------------------------------------------------
// Fast GEMM: C[M,N] = A[M,K](row stride lda, even) * B[K,N] + bias.
// Requires M%16==0 and (16*NT) | N.  One wave32 computes a 16 x (16*NT)
// strip: per k-step one b64 A-load feeds NT independent
// v_wmma_f32_16x16x4_f32 ops (XDL pipelining + NTx A-traffic reuse).
// Main k-loop is software-pipelined (next-step fragments prefetched while
// current WMMAs issue); K%4 tail uses clamped-index + value-mask loads.
// Fragment layouts per CDNA5 ISA:
//   A (16x4):  lane l, vgpr j -> M = l%16, K = 2*(l/16)+j
//   B (4x16):  lane l, vgpr j -> N = l%16, K = 2*(l/16)+j
//   C (16x16): lane l, vgpr v -> M = v + 8*(l/16), N = l%16
// ---------------------------------------------------------------------------
template <int NT>
__global__ void gemm_bias_wmma_fast(const float* __restrict__ A,
                                    const float* __restrict__ Bm,
                                    const float* __restrict__ bias,
                                    float* __restrict__ C,
                                    int M, int N, int K, int lda) {
  const int ntM = M >> 4;
  const int ntNB = N / (16 * NT);
  const int nblk = ntM * ntNB;
  const int wid = blockIdx.x * (blockDim.x >> 5) + (threadIdx.x >> 5);
  if (wid >= nblk) return;                    // wave-uniform exit
  const int tm = (wid / ntNB) << 4;
  const int tnb = (wid % ntNB) * (16 * NT);
  const int lane = threadIdx.x & 31;
  const int half = lane >> 4;
  const int l16 = lane & 15;

  v8f acc[NT];
#pragma unroll
  for (int t = 0; t < NT; ++t) acc[t] = (v8f){};

  const float* Arow = A + (size_t)(tm + l16) * lda + 2 * half;  // 8B aligned
  const float* Bcol = Bm + (size_t)(2 * half) * N + tnb + l16;
  const int K4 = K & ~3;                       // K >= 6 always -> K4 >= 4

  // ---- software-pipelined main loop over full k-steps ----
  v2f a_cur = *(const v2f*)(Arow);
  v2f b_cur[NT];
#pragma unroll
  for (int t = 0; t < NT; ++t) {
    b_cur[t][0] = Bcol[16 * t];
    b_cur[t][1] = Bcol[16 * t + N];
  }
  for (int k0 = 4; k0 < K4; k0 += 4) {
    v2f a_nxt = *(const v2f*)(Arow + k0);      // prefetch next step
    v2f b_nxt[NT];
    const float* Bp = Bcol + (size_t)k0 * N;
#pragma unroll
    for (int t = 0; t < NT; ++t) {
      b_nxt[t][0] = Bp[16 * t];
      b_nxt[t][1] = Bp[16 * t + N];
    }
#pragma unroll
    for (int t = 0; t < NT; ++t)               // NT independent WMMAs
      acc[t] = __builtin_amdgcn_wmma_f32_16x16x4_f32(false, a_cur, false,
                                                     b_cur[t], (short)0,
                                                     acc[t], false, false);
    a_cur = a_nxt;
#pragma unroll
    for (int t = 0; t < NT; ++t) b_cur[t] = b_nxt[t];
  }
#pragma unroll
  for (int t = 0; t < NT; ++t)                 // epilogue full step
    acc[t] = __builtin_amdgcn_wmma_f32_16x16x4_f32(false, a_cur, false,
                                                   b_cur[t], (short)0, acc[t],
                                                   false, false);

  if (K4 < K) {                                // masked tail, branch-free
    v2f a, b[NT];
#pragma unroll
    for (int j = 0; j < 2; ++j) {
      const int kk = K4 + 2 * half + j;
      const int kc = kk < K ? kk : K - 1;      // safe address
      const float msk = kk < K ? 1.0f : 0.0f;
      a[j] = Arow[kc - 2 * half] * msk;
      const float* Bp = Bm + (size_t)kc * N + tnb + l16;
#pragma unroll
      for (int t = 0; t < NT; ++t) b[t][j] = Bp[16 * t] * msk;
    }
#pragma unroll
    for (int t = 0; t < NT; ++t)
      acc[t] = __builtin_amdgcn_wmma_f32_16x16x4_f32(false, a, false, b[t],
                                                     (short)0, acc[t], false,
                                                     false);
  }
#pragma unroll
  for (int t = 0; t < NT; ++t) {
    const int col = tnb + 16 * t + l16;
    const float bb = bias ? bias[col] : 0.0f;
#pragma unroll
    for (int v = 0; v < 8; ++v)
      C[(size_t)(tm + v + 8 * half) * N + col] = acc[t][v] + bb;
  }
}

// ---------------------------------------------------------------------------
// Generic guarded GEMM (used for the M=8 head layers).  B(k,n) = Bm[n*K+k]
// when TRANS_B (x @ W.T with W stored (N,K) row-major).  Loads use
// clamped-index + value-mask so no EXEC divergence in the k-loop.
// ---------------------------------------------------------------------------
template <bool TRANS_B>
__global__ void gemm_bias_wmma_gen(const float* __restrict__ A,
                                   const float* __restrict__ Bm,
                                   const float* __restrict__ bias,
                                   float* __restrict__ C,
                                   int M, int N, int K, int lda) {
  const int ntN = (N + 15) >> 4;
  const int ntM = (M + 15) >> 4;
  const int ntiles = ntN * ntM;
  const int tile = blockIdx.x * (blockDim.x >> 5) + (threadIdx.x >> 5);
  if (tile >= ntiles) return;
  const int tm = (tile / ntN) << 4;
  const int tn = (tile % ntN) << 4;
  const int lane = threadIdx.x & 31;
  const int half = lane >> 4;
  const int l16 = lane & 15;
  const int arow = tm + l16;
  const int bcol = tn + l16;
  const int arc = arow < M ? arow : M - 1;
  const int bcc = bcol < N ? bcol : N - 1;
  const float am = arow < M ? 1.0f : 0.0f;
  const float bm = bcol < N ? 1.0f : 0.0f;
  v8f acc = {};
  for (int k0 = 0; k0 < K; k0 += 4) {
    v2f a, b;
#pragma unroll
    for (int j = 0; j < 2; ++j) {
      const int kk = k0 + 2 * half + j;
      const int kc = kk < K ? kk : K - 1;
      const float km = kk < K ? 1.0f : 0.0f;
      a[j] = A[(size_t)arc * lda + kc] * (am * km);
      const float bv = TRANS_B ? Bm[(size_t)bcc * K + kc]
                               : Bm[(size_t)kc * N + bcc];
      b[j] = bv * (bm * km);
    }
    acc = __builtin_amdgcn_wmma_f32_16x16x4_f32(false, a, false, b, (short)0,
                                                acc, false, false);
  }
  if (bcol < N) {
    const float bb = bias ? bias[bcol] : 0.0f;
#pragma unroll
    for (int v = 0; v < 8; ++v) {
      const int row = tm + v + 8 * half;
      if (row < M) C[(size_t)row * N + bcol] = acc[v] + bb;
    }
  }
}

// ---------------------------------------------------------------------------
// Split (B,6,N) -> xyz (B,N,3), norm (B,N,3)
// ---------------------------------------------------------------------------
__global__ void split_transpose(const float* __restrict__ in,
                                float* __restrict__ xyz,
                                float* __restrict__ nrm, int B, int N) {
  size_t t = (size_t)blockIdx.x * blockDim.x + threadIdx.x;
  size_t total = (size_t)B * N * 3;
  if (t >= total) return;
  int c = (int)(t % 3);
  int n = (int)((t / 3) % N);
  int b = (int)(t / (3 * (size_t)N));
  xyz[t] = in[((size_t)b * 6 + c) * N + n];
  nrm[t] = in[((size_t)b * 6 + c + 3) * N + n];
}

// ---------------------------------------------------------------------------
// Farthest point sampling: one block (256 threads) per batch, LDS dist array.
// ---------------------------------------------------------------------------
__global__ void fps_kernel(const float* __restrict__ xyz, int* __restrict__ fidx,
                           int npoint, int N) {
  const int b = blockIdx.x;
  __shared__ float dist[2048];
  __shared__ float redv[256];
  __shared__ int redi[256];
  __shared__ int s_far;
  const float* px = xyz + (size_t)b * N * 3;
  for (int i = threadIdx.x; i < N; i += 256) dist[i] = 1e10f;
  if (threadIdx.x == 0) s_far = 0;
  __syncthreads();
  for (int s = 0; s < npoint; ++s) {
    const int far = s_far;
    if (threadIdx.x == 0) fidx[b * npoint + s] = far;
    const float cx = px[far * 3 + 0], cy = px[far * 3 + 1], cz = px[far * 3 + 2];
    float best = -1.0f;
    int bi = 0;
    for (int i = threadIdx.x; i < N; i += 256) {
      float dx = px[i * 3 + 0] - cx;
      float dy = px[i * 3 + 1] - cy;
      float dz = px[i * 3 + 2] - cz;
      float d = dx * dx + dy * dy + dz * dz;
      float m = fminf(dist[i], d);
      dist[i] = m;
      if (m > best) { best = m; bi = i; }
    }
    redv[threadIdx.x] = best;
    redi[threadIdx.x] = bi;
    __syncthreads();
    for (int off = 128; off > 0; off >>= 1) {
      if (threadIdx.x < off) {
        float v2 = redv[threadIdx.x + off];
        int i2 = redi[threadIdx.x + off];
        if (v2 > redv[threadIdx.x] ||
            (v2 == redv[threadIdx.x] && i2 < redi[threadIdx.x])) {
          redv[threadIdx.x] = v2;
          redi[threadIdx.x] = i2;
        }
      }
      __syncthreads();
    }
    if (threadIdx.x == 0) s_far = redi[0];
    __syncthreads();
  }
}

// out[b,s,c] = pts[b, idx[b,s], c]
__global__ void gather_points(const float* __restrict__ pts,
                              const int* __restrict__ idx,
                              float* __restrict__ out, int B, int N, int S,
                              int C) {
  size_t t = (size_t)blockIdx.x * blockDim.x + threadIdx.x;
  size_t total = (size_t)B * S * C;
  if (t >= total) return;
  int c = (int)(t % C);
  int s = (int)((t / C) % S);
  int b = (int)(t / ((size_t)C * S));
  out[t] = pts[((size_t)b * N + idx[b * S + s]) * C + c];
}

// First-K indices (index order) with dist <= r^2; pad with first hit.
__global__ void query_ball(const float* __restrict__ xyz,
                           const float* __restrict__ new_xyz,
                           int* __restrict__ gidx, int B, int N, int S, int K,
                           float r2) {
  int t = blockIdx.x * blockDim.x + threadIdx.x;
  if (t >= B * S) return;
  int b = t / S;
  const float* px = xyz + (size_t)b * N * 3;
  float cx = new_xyz[t * 3 + 0], cy = new_xyz[t * 3 + 1], cz = new_xyz[t * 3 + 2];
  int* out = gidx + (size_t)t * K;
  int cnt = 0, first = 0;
  bool found = false;
  for (int i = 0; i < N && cnt < K; ++i) {
    float dx = px[i * 3 + 0] - cx;
    float dy = px[i * 3 + 1] - cy;
    float dz = px[i * 3 + 2] - cz;
    if (dx * dx + dy * dy + dz * dz <= r2) {
      if (!found) { first = i; found = true; }
      out[cnt++] = i;
    }
  }
  for (; cnt < K; ++cnt) out[cnt] = first;
}

// feat row stride Cpad (>= Cp+3, even); pad columns zero-filled.
// feat[row, :] = [ points[b,gidx,:Cp], xyz[b,gidx,:]-new_xyz[b,s,:], 0.. ]
__global__ void group_feat(const float* __restrict__ xyz,
                           const float* __restrict__ new_xyz,
                           const float* __restrict__ points,
                           const int* __restrict__ gidx,
                           float* __restrict__ feat, int B, int N, int S, int K,
                           int Cp, int Cpad) {
  const int C = Cp + 3;
  size_t t = (size_t)blockIdx.x * blockDim.x + threadIdx.x;
  size_t total = (size_t)B * S * K;
  if (t >= total) return;
  int s = (int)((t / K) % S);
  int b = (int)(t / ((size_t)K * S));
  int gi = gidx[t];
  float* o = feat + t * (size_t)Cpad;
  if (points) {
    const float* p = points + ((size_t)b * N + gi) * Cp;
    for (int c = 0; c < Cp; ++c) o[c] = p[c];
  }
  const float* g = xyz + ((size_t)b * N + gi) * 3;
  const float* nz = new_xyz + ((size_t)b * S + s) * 3;
  o[Cp + 0] = g[0] - nz[0];
  o[Cp + 1] = g[1] - nz[1];
  o[Cp + 2] = g[2] - nz[2];
  for (int c = C; c < Cpad; ++c) o[c] = 0.0f;
}

// Per-channel mean / population variance over M rows (one block per channel).
__global__ void bn_stats(const float* __restrict__ x, float* __restrict__ mean,
                         float* __restrict__ var, size_t M, int C) {
  const int c = blockIdx.x;
  float s = 0.0f, ss = 0.0f;
  for (size_t i = threadIdx.x; i < M; i += 256) {
    float v = x[i * C + c];
    s += v;
    ss += v * v;
  }
  __shared__ float sh1[256], sh2[256];
  sh1[threadIdx.x] = s;
  sh2[threadIdx.x] = ss;
  __syncthreads();
  for (int off = 128; off > 0; off >>= 1) {
    if (threadIdx.x < off) {
      sh1[threadIdx.x] += sh1[threadIdx.x + off];
      sh2[threadIdx.x] += sh2[threadIdx.x + off];
    }
    __syncthreads();
  }
  if (threadIdx.x == 0) {
    float m = sh1[0] / (float)M;
    mean[c] = m;
    var[c] = sh2[0] / (float)M - m * m;
  }
}

__global__ void bn_relu(float* __restrict__ x, const float* __restrict__ mean,
                        const float* __restrict__ var,
                        const float* __restrict__ g,
                        const float* __restrict__ be, size_t total, int C) {
  size_t t = (size_t)blockIdx.x * blockDim.x + threadIdx.x;
  if (t >= total) return;
  int c = (int)(t % C);
  float v = (x[t] - mean[c]) * rsqrtf(var[c] + BN_EPS) * g[c] + be[c];
  x[t] = fmaxf(v, 0.0f);
}

// out[bs*Ctot + coff + c] = max_k feat[(bs*K+k)*C + c]
__global__ void maxpool_concat(const float* __restrict__ feat,
                               float* __restrict__ out, int B, int S, int K,
                               int C, int Ctot, int coff) {
  size_t t = (size_t)blockIdx.x * blockDim.x + threadIdx.x;
  size_t total = (size_t)B * S * C;
  if (t >= total) return;
  int c = (int)(t % C);
  size_t bs = t / C;
  const float* f = feat + bs * (size_t)K * C + c;
  float m = -INFINITY;
  for (int k = 0; k < K; ++k) m = fmaxf(m, f[(size_t)k * C]);
  out[bs * Ctot + coff + c] = m;
}

// out row stride Cpad; [xyz(3), pts(Cp), zero-pad]
__global__ void concat_xyz_pts(const float* __restrict__ xyz,
                               const float* __restrict__ pts,
                               float* __restrict__ out, int BS, int Cp,
                               int Cpad) {
  size_t t = (size_t)blockIdx.x * blockDim.x + threadIdx.x;
  size_t total = (size_t)BS * Cpad;
  if (t >= total) return;
  int c = (int)(t % Cpad);
  size_t bs = t / Cpad;
  float v = 0.0f;
  if (c < 3) v = xyz[bs * 3 + c];
  else if (c < 3 + Cp) v = pts[bs * (size_t)Cp + (c - 3)];
  out[t] = v;
}

// BN over axis 0 (tiny M rows) + relu, one thread per channel.
__global__ void bn1d_relu(float* __restrict__ x, const float* __restrict__ g,
                          const float* __restrict__ be, int M, int C) {
  int c = blockIdx.x * blockDim.x + threadIdx.x;
  if (c >= C) return;
  float s = 0.0f, ss = 0.0f;
  for (int i = 0; i < M; ++i) {
    float v = x[i * C + c];
    s += v;
    ss += v * v;
  }
  float m = s / (float)M;
  float inv = rsqrtf(ss / (float)M - m * m + BN_EPS);
  for (int i = 0; i < M; ++i) {
    float v = (x[i * C + c] - m) * inv * g[c] + be[c];
    x[i * C + c] = fmaxf(v, 0.0f);
  }
}

__global__ void log_softmax_rows(const float* __restrict__ in,
                                 float* __restrict__ out, int R, int C) {
  int r = blockIdx.x * blockDim.x + threadIdx.x;
  if (r >= R) return;
  const float* x = in + (size_t)r * C;
  float m = -INFINITY;
  for (int c = 0; c < C; ++c) m = fmaxf(m, x[c]);
  float s = 0.0f;
  for (int c = 0; c < C; ++c) s += expf(x[c] - m);
  float ls = m + logf(s);
  float* o = out + (size_t)r * C;
  for (int c = 0; c < C; ++c) o[c] = x[c] - ls;
}

__global__ void logsumexp5(const float* __restrict__ lb, float* __restrict__ out,
                           int T) {
  int t = blockIdx.x * blockDim.x + threadIdx.x;
  if (t >= T) return;
  float m = -INFINITY;
  for (int j = 0; j < 5; ++j) m = fmaxf(m, lb[j * T + t]);
  float s = 0.0f;
  for (int j = 0; j < 5; ++j) s += expf(lb[j * T + t] - m);
  out[t] = m + logf(s);
}

// out[(c1*J + j)*C2 + c2] = W[c1*C2 + c2]   ((J,C1,C2) -> (C1,J,C2) permute)
__global__ void stack_permute(const float* __restrict__ W,
                              float* __restrict__ out, int C1, int C2, int j,
                              int J) {
  size_t t = (size_t)blockIdx.x * blockDim.x + threadIdx.x;
  size_t total = (size_t)C1 * C2;
  if (t >= total) return;
  int c1 = (int)(t / C2);
  int c2 = (int)(t % C2);
  out[((size_t)c1 * J + j) * C2 + c2] = W[t];
}

// ---------------------------------------------------------------------------
struct ConvParams { const float *W, *b, *g, *be; };

static void launch_gemm(const float* A, const float* W, const float* bias,
                        float* out, int M, int N, int K, int lda, bool transB,
                        hipStream_t stream) {
  if (!transB && (M % 16 == 0) && (lda % 2 == 0) && (N % 32 == 0)) {
    if (N % 64 == 0) {
      int nblk = (M / 16) * (N / 64);
      gemm_bias_wmma_fast<4><<<cdiv(nblk, 8), 256, 0, stream>>>(A, W, bias, out,
                                                                M, N, K, lda);
    } else if (N % 96 == 0) {
      int nblk = (M / 16) * (N / 96);
      gemm_bias_wmma_fast<6><<<cdiv(nblk, 8), 256, 0, stream>>>(A, W, bias, out,
                                                                M, N, K, lda);
    } else {
      int nblk = (M / 16) * (N / 32);
      gemm_bias_wmma_fast<2><<<cdiv(nblk, 8), 256, 0, stream>>>(A, W, bias, out,
                                                                M, N, K, lda);
    }
  } else if (transB) {
    int ntiles = cdiv(M, 16) * cdiv(N, 16);
    gemm_bias_wmma_gen<true><<<cdiv(ntiles, 8), 256, 0, stream>>>(
        A, W, bias, out, M, N, K, lda);
  } else {
    int ntiles = cdiv(M, 16) * cdiv(N, 16);
    gemm_bias_wmma_gen<false><<<cdiv(ntiles, 8), 256, 0, stream>>>(
        A, W, bias, out, M, N, K, lda);
  }
}

static void run_sa_msg(const float* xyz, const float* pts, int Cp, int B, int N,
                       int S, const float* radii, const int* Ks,
                       const ConvParams pr[3][3], const int chs[3][4], int* fidx,
                       int* gidx, float* new_xyz, float* out_pts, int Ctot,
                       float* bufA, float* bufB, float* mean, float* var,
                       hipStream_t stream) {
  fps_kernel<<<B, 256, 0, stream>>>(xyz, fidx, S, N);
  gather_points<<<grid1((size_t)B * S * 3), 256, 0, stream>>>(xyz, fidx, new_xyz,
                                                              B, N, S, 3);
  int coff = 0;
  for (int br = 0; br < 3; ++br) {
    const int K = Ks[br];
    const float r2 = radii[br] * radii[br];
    query_ball<<<grid1((size_t)B * S), 256, 0, stream>>>(xyz, new_xyz, gidx, B,
                                                         N, S, K, r2);
    const size_t rows = (size_t)B * S * K;
    const int C0 = chs[br][0];
    const int C0pad = (C0 + 1) & ~1;  // even row stride for b64 A-loads
    group_feat<<<grid1(rows), 256, 0, stream>>>(xyz, new_xyz, pts, gidx, bufA, B,
                                                N, S, K, Cp, C0pad);
    float* cur = bufA;
    float* nxt = bufB;
    int cin = C0;
    int lda = C0pad;
    for (int l = 0; l < 3; ++l) {
      const int cout = chs[br][l + 1];
      launch_gemm(cur, pr[br][l].W, pr[br][l].b, nxt, (int)rows, cout, cin, lda,
                  false, stream);
      bn_stats<<<cout, 256, 0, stream>>>(nxt, mean, var, rows, cout);
      bn_relu<<<grid1(rows * cout), 256, 0, stream>>>(nxt, mean, var,
                                                      pr[br][l].g, pr[br][l].be,
                                                      rows * cout, cout);
      float* t = cur; cur = nxt; nxt = t;
      cin = cout;
      lda = cout;
    }
    maxpool_concat<<<grid1((size_t)B * S * cin), 256, 0, stream>>>(
        cur, out_pts, B, S, K, cin, Ctot, coff);
    coff += cin;
  }
}

// ---------------------------------------------------------------------------
extern "C" void kernel_launch(void* const* d_in, const int* in_sizes, int n_in,
                              void* d_out, int out_size, void* d_ws,
                              size_t ws_size, hipStream_t stream) {
  (void)in_sizes; (void)n_in; (void)out_size; (void)ws_size;
  const int B = 8;

  // ---- input mapping: [xyz] + pytree leaves (sorted dict keys) ----
  int p = 0;
  const float* xyz_in = (const float*)d_in[p++];
  const float* headp[6][10];  // b1..b5, g2 ; (W1,b1,g1,be1,W2,b2,g2,be2,W3,b3)
  for (int h = 0; h < 6; ++h)
    for (int i = 0; i < 10; ++i) headp[h][i] = (const float*)d_in[p++];
  ConvParams sa1[3][3], sa2[3][3], sa3l[3];
  for (int br = 0; br < 3; ++br)
    for (int l = 0; l < 3; ++l) {
      sa1[br][l] = {(const float*)d_in[p], (const float*)d_in[p + 1],
                    (const float*)d_in[p + 2], (const float*)d_in[p + 3]};
      p += 4;
    }
  for (int br = 0; br < 3; ++br)
    for (int l = 0; l < 3; ++l) {
      sa2[br][l] = {(const float*)d_in[p], (const float*)d_in[p + 1],
                    (const float*)d_in[p + 2], (const float*)d_in[p + 3]};
      p += 4;
    }
  for (int l = 0; l < 3; ++l) {
    sa3l[l] = {(const float*)d_in[p], (const float*)d_in[p + 1],
               (const float*)d_in[p + 2], (const float*)d_in[p + 3]};
    p += 4;
  }

  // ---- workspace bump allocator ----
  char* ws = (char*)d_ws;
  size_t off = 0;
  auto alloc_f = [&](size_t nfl) -> float* {
    float* q = (float*)(ws + off);
    off = (off + nfl * sizeof(float) + 255) & ~(size_t)255;
    return q;
  };
  auto alloc_i = [&](size_t ni) -> int* {
    int* q = (int*)(ws + off);
    off = (off + ni * sizeof(int) + 255) & ~(size_t)255;
    return q;
  };
  float* f_xyz  = alloc_f((size_t)B * 2048 * 3);
  float* f_norm = alloc_f((size_t)B * 2048 * 3);
  float* l1_xyz = alloc_f((size_t)B * 512 * 3);
  float* l1_pts = alloc_f((size_t)B * 512 * 320);
  float* l2_xyz = alloc_f((size_t)B * 128 * 3);
  float* l2_pts = alloc_f((size_t)B * 128 * 640);
  float* mean   = alloc_f(1024);
  float* var    = alloc_f(1024);
  float* x_feat = alloc_f((size_t)B * 1024);
  float* h1     = alloc_f((size_t)B * 512);
  float* h2     = alloc_f((size_t)B * 256);
  float* logits = alloc_f((size_t)B * 40);
  float* lb     = alloc_f((size_t)5 * B * 40);
  int* fidx     = alloc_i((size_t)B * 512);
  int* gidx     = alloc_i((size_t)B * 512 * 128);
  float* bufA   = alloc_f(50331648);  // max: 524288x96 / 131072x324 rows
  float* bufB   = alloc_f(67108864);  // max: 524288x128

  // ---- output offsets (floats) ----
  float* out = (float*)d_out;
  const size_t off_xg2 = 0;
  const size_t off_l3  = off_xg2 + 8 * 40;
  const size_t off_xb  = off_l3 + 8 * 1024;
  const size_t off_g0  = off_xb + 8 * 40;
  const size_t off_g4  = off_g0 + 512 * 1024;
  const size_t off_g8  = off_g4 + 256 * 512;
  const size_t off_b0  = off_g8 + 40 * 256;
  const size_t off_b4  = off_b0 + (size_t)512 * 5 * 1024;
  const size_t off_b8  = off_b4 + (size_t)256 * 5 * 512;

  // ---- stage 0: split input ----
  split_transpose<<<grid1((size_t)B * 2048 * 3), 256, 0, stream>>>(
      xyz_in, f_xyz, f_norm, B, 2048);

  // ---- SA1 ----
  {
    const float radii[3] = {0.1f, 0.2f, 0.4f};
    const int Ks[3] = {16, 32, 128};
    const int chs[3][4] = {{6, 32, 32, 64}, {6, 64, 64, 128}, {6, 64, 96, 128}};
    run_sa_msg(f_xyz, f_norm, 3, B, 2048, 512, radii, Ks, sa1, chs, fidx, gidx,
               l1_xyz, l1_pts, 320, bufA, bufB, mean, var, stream);
  }
  // ---- SA2 ----
  {
    const float radii[3] = {0.2f, 0.4f, 0.8f};
    const int Ks[3] = {32, 64, 128};
    const int chs[3][4] = {{323, 64, 64, 128},
                           {323, 128, 128, 256},
                           {323, 128, 128, 256}};
    run_sa_msg(l1_xyz, l1_pts, 320, B, 512, 128, radii, Ks, sa2, chs, fidx,
               gidx, l2_xyz, l2_pts, 640, bufA, bufB, mean, var, stream);
  }
  // ---- SA3: concat [l2_xyz, l2_pts] -> (B*128, 643 pad 644), 3 convs, pool
  {
    const int BS = B * 128;
    const int C0 = 643, C0pad = 644;
    concat_xyz_pts<<<grid1((size_t)BS * C0pad), 256, 0, stream>>>(
        l2_xyz, l2_pts, bufA, BS, 640, C0pad);
    const int chs[4] = {C0, 256, 512, 1024};
    float* cur = bufA;
    float* nxt = bufB;
    int lda = C0pad;
    for (int l = 0; l < 3; ++l) {
      launch_gemm(cur, sa3l[l].W, sa3l[l].b, nxt, BS, chs[l + 1], chs[l], lda,
                  false, stream);
      bn_stats<<<chs[l + 1], 256, 0, stream>>>(nxt, mean, var, (size_t)BS,
                                               chs[l + 1]);
      bn_relu<<<grid1((size_t)BS * chs[l + 1]), 256, 0, stream>>>(
          nxt, mean, var, sa3l[l].g, sa3l[l].be, (size_t)BS * chs[l + 1],
          chs[l + 1]);
      float* t = cur; cur = nxt; nxt = t;
      lda = chs[l + 1];
    }
    // max over the 128 grouped points -> (B, 1024) == l3_points flat
    maxpool_concat<<<grid1((size_t)B * 1024), 256, 0, stream>>>(
        cur, x_feat, B, 1, 128, 1024, 1024, 0);
  }
  hipMemcpyAsync(out + off_l3, x_feat, (size_t)B * 1024 * sizeof(float),
                 hipMemcpyDeviceToDevice, stream);

  // ---- heads (sorted order: b1..b5 = 0..4, g2 = 5) ----
  for (int h = 0; h < 6; ++h) {
    const float* const* P = headp[h];
    launch_gemm(x_feat, P[0], P[1], h1, B, 512, 1024, 1024, true, stream);
    bn1d_relu<<<grid1(512), 256, 0, stream>>>(h1, P[2], P[3], B, 512);
    launch_gemm(h1, P[4], P[5], h2, B, 256, 512, 512, true, stream);
    bn1d_relu<<<grid1(256), 256, 0, stream>>>(h2, P[6], P[7], B, 256);
    launch_gemm(h2, P[8], P[9], logits, B, 40, 256, 256, true, stream);
    float* dst = (h == 5) ? (out + off_xg2) : (lb + (size_t)h * B * 40);
    log_softmax_rows<<<1, 32, 0, stream>>>(logits, dst, B, 40);
  }
  logsumexp5<<<grid1((size_t)B * 40), 256, 0, stream>>>(lb, out + off_xb,
                                                        B * 40);

  // ---- weight copy / permute outputs ----
  hipMemcpyAsync(out + off_g0, headp[5][0], (size_t)512 * 1024 * sizeof(float),
                 hipMemcpyDeviceToDevice, stream);
  hipMemcpyAsync(out + off_g4, headp[5][4], (size_t)256 * 512 * sizeof(float),
                 hipMemcpyDeviceToDevice, stream);
  hipMemcpyAsync(out + off_g8, headp[5][8], (size_t)40 * 256 * sizeof(float),
                 hipMemcpyDeviceToDevice, stream);
  for (int j = 0; j < 5; ++j) {
    stack_permute<<<grid1((size_t)512 * 1024), 256, 0, stream>>>(
        headp[j][0], out + off_b0, 512, 1024, j, 5);
    stack_permute<<<grid1((size_t)256 * 512), 256, 0, stream>>>(
        headp[j][4], out + off_b4, 256, 512, j, 5);
    stack_permute<<<grid1((size_t)40 * 256), 256, 0, stream>>>(
        headp[j][8], out + off_b8, 40, 256, j, 5);
  }
}